// Attention_9844065042780
// MI455X (gfx1250) — compile-verified
//
#include <hip/hip_runtime.h>
#include <hip/hip_bf16.h>

typedef float v2f __attribute__((ext_vector_type(2)));
typedef float v8f __attribute__((ext_vector_type(8)));

#define BATCH 32
#define NCHN  256          // N (channels)
#define HW    196          // 14*14
#define KH    8            // K heads
#define NCLS  1000         // num classes
#define RTOT  (NCHN*HW)    // 50176, reduction length of big GEMM
#define NCHUNK 16          // split-K chunks for big GEMM
#define CHUNK (RTOT/NCHUNK) // 3136 (= 16 n-slices; 196 % 4 == 0 so 4-step never crosses n)
#define SUBR  392          // r staged per LDS pass (2 n-slices)
#define NSUB  (CHUNK/SUBR) // 8
#define YPAD  396          // LDS row stride (dwords): 396 mod 64 = 12 -> conflict-free b64 reads

// ---------------------------------------------------------------------------
// Kernel 1: ah = softmax_k(conv(x)), plus orthogonality loss per batch.
// ---------------------------------------------------------------------------
__global__ __launch_bounds__(256) void attn_softmax_loss(
    const float* __restrict__ x,       // [B][N][HW]
    const float* __restrict__ conv_w,  // [K][N]
    float* __restrict__ ah,            // out: [B][K][HW]
    float* __restrict__ loss_out)      // out: [B]
{
  __shared__ float cws[KH * NCHN];   // 8 KB
  __shared__ float ahs[KH * HW];     // 6.3 KB
  __shared__ float red[256];
  const int b = blockIdx.x;
  const int t = threadIdx.x;

  for (int i = t; i < KH * NCHN; i += 256) cws[i] = conv_w[i];
  __syncthreads();

  if (t < HW) {
    float acc[KH];
#pragma unroll
    for (int k = 0; k < KH; ++k) acc[k] = 0.f;
    const float* xb = x + (size_t)b * RTOT + t;
    for (int n = 0; n < NCHN; ++n) {
      float xv = xb[(size_t)n * HW];
#pragma unroll
      for (int k = 0; k < KH; ++k) acc[k] += xv * cws[k * NCHN + n];
    }
    float m = acc[0];
#pragma unroll
    for (int k = 1; k < KH; ++k) m = fmaxf(m, acc[k]);
    float s = 0.f;
#pragma unroll
    for (int k = 0; k < KH; ++k) { acc[k] = __expf(acc[k] - m); s += acc[k]; }
    float inv = 1.f / s;
#pragma unroll
    for (int k = 0; k < KH; ++k) {
      float a = acc[k] * inv;
      ahs[k * HW + t] = a;
      ah[((size_t)b * KH + k) * HW + t] = a;
    }
  }
  __syncthreads();

  // loss1 contribution: thread t<196 owns pair (w,v); G[w,v] = sum_{k,h} ah[k,h,w]*ah[k,h,v]
  float p = 0.f;
  if (t < HW) {
    int w = t / 14, v = t % 14;
    float G = 0.f;
    for (int k = 0; k < KH; ++k)
      for (int h = 0; h < 14; ++h)
        G += ahs[k * HW + h * 14 + w] * ahs[k * HW + h * 14 + v];
    p = G * G;
  }
  // loss2 contribution: thread t<112 owns (k,w); s = sum_h ah[k,h,w]
  if (t < KH * 14) {
    int k = t / 14, w = t % 14;
    float sh = 0.f;
    for (int h = 0; h < 14; ++h) sh += ahs[k * HW + h * 14 + w];
    p -= sh * sh;
  }
  red[t] = p;
  __syncthreads();
  for (int off = 128; off > 0; off >>= 1) {
    if (t < off) red[t] += red[t + off];
    __syncthreads();
  }
  if (t == 0) loss_out[b] = red[0];
}

// ---------------------------------------------------------------------------
// Kernel 2: dim_red <- bias (broadcast over b); re-run every call.
// ---------------------------------------------------------------------------
__global__ void init_dimred(float* __restrict__ dr, const float* __restrict__ db) {
  int i = blockIdx.x * 256 + threadIdx.x;       // 65536 total = [B][K][N]
  dr[i] = db[i & (KH * NCHN - 1)];              // (b*8+k)*256+m mod 2048 = k*256+m
}

// ---------------------------------------------------------------------------
// Kernel 3: big split-K GEMM with V_WMMA_F32_16X16X4_F32.
//   dim_red[b,k,m] += sum_r (ah[b,k,p]*x[b,n,p]) * dimred_w[k,m,r],  r=(n,p)
// Grid: 8 k * 16 chunks; block = 1024 threads = 32 waves; each wave owns one
// 16x16 C tile (btile 0/1  x  mtile 0..15). A staged via LDS; the 411 MB W
// stream is loaded non-temporal (TH=NT) so it does not thrash the 192 MB L2
// that holds the reused x/ah/dim_red tensors.
// ---------------------------------------------------------------------------
__global__ __launch_bounds__(1024) void dimred_gemm(
    const float* __restrict__ dw,   // [K][256][RTOT]
    const float* __restrict__ ah,   // [B][K][HW]
    const float* __restrict__ x,    // [B][RTOT]
    float* __restrict__ dr)         // [B][K][256] (pre-biased)
{
  __shared__ float ys[BATCH * YPAD];            // 50,688 B

  const int k     = blockIdx.x / NCHUNK;
  const int chunk = blockIdx.x % NCHUNK;
  const int t     = threadIdx.x;
  const int wv    = t >> 5;
  const int lane  = t & 31;
  const int btile = wv & 1;          // b rows 0-15 / 16-31
  const int mtile = wv >> 1;         // m cols, 16 tiles
  const int lrow  = lane & 15;
  const int ldel  = (lane >> 4) * 2; // K-halves per WMMA f32 A/B layout

  const size_t rbase = (size_t)chunk * CHUNK;
  const float* wk = dw + (size_t)k * NCHN * RTOT + (size_t)(mtile * 16 + lrow) * RTOT;
  const float* yrow = &ys[(btile * 16 + lrow) * YPAD + ldel];

  v8f acc = {};
  for (int sub = 0; sub < NSUB; ++sub) {
    const size_t r0 = rbase + (size_t)sub * SUBR;
    __syncthreads();
    // stage y[b, rr] = ah[b,k,p] * x[b, r0+rr]  (coalesced in rr)
    for (int i = t; i < BATCH * SUBR; i += 1024) {
      int bb = i / SUBR;
      int rr = i - bb * SUBR;
      float xv = x[(size_t)bb * RTOT + r0 + rr];
      float av = ah[((size_t)bb * KH + k) * HW + (rr % HW)];   // r0 is a multiple of HW
      ys[bb * YPAD + rr] = xv * av;
    }
    __syncthreads();

    const v2f* wp = (const v2f*)(wk + r0 + ldel);
    for (int s = 0; s < SUBR / 4; ++s) {
      v2f a = *(const v2f*)(yrow + 4 * s);            // ds_load_b64, conflict-free
      v2f w = __builtin_nontemporal_load(wp + 2 * s); // global_load_b64 TH=NT stream
      acc = __builtin_amdgcn_wmma_f32_16x16x4_f32(
          false, a, false, w, (short)0, acc, false, false);
    }
  }

  // C/D layout: VGPR j, lane -> M = j + 8*(lane>=16), N = lane%16
  const int mcol = mtile * 16 + lrow;
  const int moff = (lane >> 4) * 8;
#pragma unroll
  for (int j = 0; j < 8; ++j) {
    int bb = btile * 16 + j + moff;
    unsafeAtomicAdd(&dr[((size_t)bb * KH + k) * NCHN + mcol], acc[j]);
  }
}

// ---------------------------------------------------------------------------
// Kernel 4: hyp[b,k,c] = sum_n dim_red[b,k,n]*Wo_w[k,c,n] + Wo_b[k,c] (WMMA)
// Grid: 8 k * 16 groups; block = 128 threads = 4 waves; wave -> one c-tile.
// ---------------------------------------------------------------------------
__global__ __launch_bounds__(128) void hyp_gemm(
    const float* __restrict__ dr,   // [B][K][256]
    const float* __restrict__ wo,   // [K][1000][256]
    const float* __restrict__ wob,  // [K][1000]
    float* __restrict__ hyp)        // [B][K][1000]
{
  const int k  = blockIdx.x >> 4;
  const int cg = blockIdx.x & 15;
  const int t  = threadIdx.x;
  const int wv = t >> 5;
  const int lane = t & 31;
  const int ctile = cg * 4 + wv;
  if (ctile >= 63) return;                 // 63 tiles cover 1000 (wave-uniform)

  const int lrow = lane & 15;
  const int ldel = (lane >> 4) * 2;
  const int c  = ctile * 16 + lrow;
  const int cl = (c < NCLS) ? c : (NCLS - 1);   // clamp tail reads, discard results

  const v2f* wp    = (const v2f*)(wo + ((size_t)k * NCLS + cl) * NCHN + ldel);
  const float* a0p = dr + ((size_t)lrow * KH + k) * NCHN + ldel;        // b rows 0-15
  const float* a1p = dr + ((size_t)(16 + lrow) * KH + k) * NCHN + ldel; // b rows 16-31

  v8f c0 = {}, c1 = {};
  for (int s = 0; s < NCHN / 4; ++s) {
    v2f bw = __builtin_nontemporal_load(wp + 2 * s);   // Wo read exactly once: NT
    v2f a0 = *(const v2f*)(a0p + 4 * s);
    v2f a1 = *(const v2f*)(a1p + 4 * s);
    c0 = __builtin_amdgcn_wmma_f32_16x16x4_f32(false, a0, false, bw, (short)0, c0, false, false);
    c1 = __builtin_amdgcn_wmma_f32_16x16x4_f32(false, a1, false, bw, (short)0, c1, false, false);
  }

  if (c < NCLS) {
    float bias = wob[k * NCLS + c];
    const int moff = (lane >> 4) * 8;
#pragma unroll
    for (int j = 0; j < 8; ++j) {
      int b0 = j + moff;
      hyp[((size_t)b0 * KH + k) * NCLS + c]        = c0[j] + bias;
      hyp[((size_t)(b0 + 16) * KH + k) * NCLS + c] = c1[j] + bias;
    }
  }
}

// ---------------------------------------------------------------------------
// Kernel 5: conf[b,k] = tanh(dim_red[b,k,:] . Wg_w[k,:] + Wg_b[k])
// ---------------------------------------------------------------------------
__global__ __launch_bounds__(256) void conf_kernel(
    const float* __restrict__ dr,
    const float* __restrict__ wg,   // [K][256]
    const float* __restrict__ wgb,  // [K]
    float* __restrict__ conf)       // [B][K]
{
  int t = threadIdx.x;              // 256 = B*K, t = b*8+k
  int k = t & 7;
  float acc = 0.f;
  const float* d = dr + (size_t)t * NCHN;
  for (int n = 0; n < NCHN; ++n) acc += d[n] * wg[k * NCHN + n];
  conf[t] = tanhf(acc + wgb[k]);
}

// ---------------------------------------------------------------------------
extern "C" void kernel_launch(void* const* d_in, const int* in_sizes, int n_in,
                              void* d_out, int out_size, void* d_ws, size_t ws_size,
                              hipStream_t stream) {
  const float* x   = (const float*)d_in[0];
  const float* cw  = (const float*)d_in[1];
  const float* dw  = (const float*)d_in[2];
  const float* db  = (const float*)d_in[3];
  const float* wo  = (const float*)d_in[4];
  const float* wob = (const float*)d_in[5];
  const float* wg  = (const float*)d_in[6];
  const float* wgb = (const float*)d_in[7];

  float* out  = (float*)d_out;
  float* hyp  = out;               // [B,K,1000] = 256000
  float* conf = out + 256000;      // [B,K,1]    = 256
  float* loss = out + 256256;      // [B,1]      = 32

  float* ws = (float*)d_ws;
  float* ah = ws;                  // [B,K,HW]   = 50176 floats
  float* dr = ws + 50176;          // [B,K,N]    = 65536 floats  (total < 0.5 MB)

  attn_softmax_loss<<<dim3(BATCH), dim3(256), 0, stream>>>(x, cw, ah, loss);
  init_dimred<<<dim3(256), dim3(256), 0, stream>>>(dr, db);
  dimred_gemm<<<dim3(KH * NCHUNK), dim3(1024), 0, stream>>>(dw, ah, x, dr);
  hyp_gemm<<<dim3(KH * 16), dim3(128), 0, stream>>>(dr, wo, wob, hyp);
  conf_kernel<<<dim3(1), dim3(256), 0, stream>>>(dr, wg, wgb, conf);
}